// MaskPropagation_3590592659792
// MI455X (gfx1250) — compile-verified
//
#include <hip/hip_runtime.h>

// ---- problem constants (from reference) ----
#define RR    6
#define PP    13          // window width
#define PSQ   169         // PP*PP
#define CC    64          // channels
#define HH    64
#define WW    64
#define BB    4
#define OBJ   16
#define DSUB  4
#define SCALE 0.125f      // 1/sqrt(64)

typedef __attribute__((ext_vector_type(2))) float v2f;
typedef __attribute__((ext_vector_type(8))) float v8f;

// ======================================================================
// Kernel 1: local correlation via V_WMMA_F32_16X16X4_F32.
// Block = (b, y, 16-pixel x-tile), 4 waves; wave w handles dy = w, w+4, ...
// For each dy: two 16x16 f32 WMMA accumulators over K=64 (16 steps of K=4)
// cover mem columns [x0-6 .. x0+25]; corr[m,dx] = C[m, m+dx].
// OOB handling: clamp coordinates, load unconditionally (immediate channel
// offsets), multiply by a 0/1 mask -> no exec-mask branches per load.
// ======================================================================
__global__ __launch_bounds__(128) void corr_wmma_kernel(
    const float* __restrict__ mem, const float* __restrict__ qf,
    float* __restrict__ wbuf)
{
  __shared__ float tiles[4][2][256];   // per-wave C-tile staging (8 KB)

  const int lane = threadIdx.x & 31;
  const int wave = threadIdx.x >> 5;
  const int x0 = blockIdx.x * 16;
  const int y  = blockIdx.y;
  const int b  = blockIdx.z;

  const int m     = lane & 15;          // query pixel within tile / B column
  const int khalf = (lane >> 4) * 2;    // K sub-pair per ISA A/B layout

  // ---- A fragments: Q[c, x0+m], persistent across all dy ----
  v2f afrag[16];
  const float* qb = qf + (((size_t)b * CC) * HH + y) * WW + x0 + m;
#pragma unroll
  for (int kk = 0; kk < 16; ++kk) {
    const int c0 = kk * 4 + khalf;
    afrag[kk].x = qb[(size_t)(c0 + 0) * HH * WW];
    afrag[kk].y = qb[(size_t)(c0 + 1) * HH * WW];
  }

  const float* memb = mem + (size_t)b * CC * HH * WW;

  for (int dyi = wave; dyi < PP; dyi += 4) {
    const int ym = y + dyi - RR;
    const bool rowok = (ym >= 0) && (ym < HH);
    const int xa  = x0 - RR + m;        // tile0 mem x
    const int xb2 = x0 + 10 + m;        // tile1 mem x

    // clamped coordinates + 0/1 masks (no divergent loads)
    const int ymc = min(max(ym, 0), HH - 1);
    const int xac = min(max(xa, 0), WW - 1);
    const int xbc = min(xb2, WW - 1);
    const float fa = (rowok && xa >= 0 && xa < WW) ? 1.0f : 0.0f;
    const float fb = (rowok && xb2 < WW) ? 1.0f : 0.0f;

    // per-lane base pointers; channel stride folds into imm offsets
    const float* pa = memb + (size_t)ymc * WW + xac;
    const float* pb = memb + (size_t)ymc * WW + xbc;

    // prefetch next dy's mem row into cache (global_prefetch_b8)
    if (dyi + 4 < PP) {
      int ymn = min(max(y + dyi + 4 - RR, 0), HH - 1);
      __builtin_prefetch(memb + ((size_t)khalf * HH + ymn) * WW + x0, 0, 1);
    }

    v8f acc0 = {0.f,0.f,0.f,0.f,0.f,0.f,0.f,0.f};
    v8f acc1 = {0.f,0.f,0.f,0.f,0.f,0.f,0.f,0.f};
#pragma unroll
    for (int kk = 0; kk < 16; ++kk) {
      const int c0 = kk * 4 + khalf;
      v2f b0, b1;
      b0.x = pa[(size_t)(c0 + 0) * HH * WW] * fa;
      b0.y = pa[(size_t)(c0 + 1) * HH * WW] * fa;
      b1.x = pb[(size_t)(c0 + 0) * HH * WW] * fb;
      b1.y = pb[(size_t)(c0 + 1) * HH * WW] * fb;
      acc0 = __builtin_amdgcn_wmma_f32_16x16x4_f32(false, afrag[kk], false, b0,
                                                   (short)0, acc0, false, false);
      acc1 = __builtin_amdgcn_wmma_f32_16x16x4_f32(false, afrag[kk], false, b1,
                                                   (short)0, acc1, false, false);
    }

    // C layout: acc[r] on lane L holds C[M = r + 8*(L>>4), N = L&15]
#pragma unroll
    for (int r = 0; r < 8; ++r) {
      const int mrow = r + 8 * (lane >> 4);
      tiles[wave][0][mrow * 16 + (lane & 15)] = acc0[r];
      tiles[wave][1][mrow * 16 + (lane & 15)] = acc1[r];
    }
    __builtin_amdgcn_wave_barrier();
    asm volatile("s_wait_dscnt 0" ::: "memory");   // LDS RAW within wave

    // 13*16 = 208 corr values; idx = dx*16 + m -> 16-float coalesced stores
    for (int t = 0; t < 7; ++t) {
      const int idx = t * 32 + lane;
      if (idx < PP * 16) {
        const int dxi = idx >> 4;        // 0..12
        const int mm  = idx & 15;
        const int n   = mm + dxi;        // 0..27
        const float v = (n < 16) ? tiles[wave][0][mm * 16 + n]
                                 : tiles[wave][1][mm * 16 + (n - 16)];
        const int p = dyi * PP + dxi;
        wbuf[(((size_t)b * PSQ + p) * HH + y) * WW + x0 + mm] = v * SCALE;
      }
    }
    __builtin_amdgcn_wave_barrier();
  }
}

// ======================================================================
// Kernel 2: per-pixel softmax over p (stride h*w -> coalesced across x),
// then add scaled flow bias, in place in wbuf.
// ======================================================================
__global__ __launch_bounds__(256) void softmax_flo_kernel(
    const float* __restrict__ flo, float* __restrict__ wbuf)
{
  const int id = blockIdx.x * blockDim.x + threadIdx.x;   // pixel id
  if (id >= BB * HH * WW) return;
  const int b  = id / (HH * WW);
  const int yx = id % (HH * WW);
  float* wp = wbuf + (size_t)b * PSQ * HH * WW + yx;

  float mx = -1e30f;
  for (int p = 0; p < PSQ; ++p) mx = fmaxf(mx, wp[(size_t)p * HH * WW]);
  float s = 0.0f;
  for (int p = 0; p < PSQ; ++p) s += __expf(wp[(size_t)p * HH * WW] - mx);
  const float inv = 1.0f / s;
  for (int p = 0; p < PSQ; ++p) {
    const float e = __expf(wp[(size_t)p * HH * WW] - mx) * inv;
    wp[(size_t)p * HH * WW] = e + flo[(size_t)p * HH * WW + yx] * SCALE;
  }
}

// ======================================================================
// Kernel 3: masked aggregation. 16x16 pixel tile per block; 28x28 halo of
// the 4x-subsampled mask for all 16 objects staged in LDS (~49 KB),
// then out[obj] = sum_p w[p] * msk[obj, y+dy, x+dx] (pure VALU; 44M MACs).
// ======================================================================
#define MPAD 788   // 28*28 = 784, padded; lane stride in inner dim is 1 word

__global__ __launch_bounds__(256) void aggregate_kernel(
    const float* __restrict__ mskmem, const float* __restrict__ wbuf,
    float* __restrict__ out)
{
  __shared__ float msk_s[OBJ * MPAD];
  const int b   = blockIdx.z;
  const int ty0 = blockIdx.y * 16, tx0 = blockIdx.x * 16;
  const int tid = threadIdx.x;

  for (int i = tid; i < OBJ * 784; i += 256) {
    const int obj = i / 784;
    const int r   = i % 784;
    const int row = r / 28, col = r % 28;
    const int gy = ty0 - RR + row, gx = tx0 - RR + col;
    float v = 0.0f;
    if (gy >= 0 && gy < HH && gx >= 0 && gx < WW)
      v = mskmem[(((size_t)b * OBJ + obj) * (HH * DSUB) + gy * DSUB)
                   * (WW * DSUB) + gx * DSUB];
    msk_s[obj * MPAD + row * 28 + col] = v;
  }
  __syncthreads();

  const int tx = tid & 15, ty = tid >> 4;
  const int y = ty0 + ty, x = tx0 + tx;
  float acc[OBJ];
#pragma unroll
  for (int o = 0; o < OBJ; ++o) acc[o] = 0.0f;

  const float* wp = wbuf + (((size_t)b * PSQ) * HH + y) * WW + x;
  for (int dyi = 0; dyi < PP; ++dyi) {
    for (int dxi = 0; dxi < PP; ++dxi) {
      const float wv = wp[(size_t)(dyi * PP + dxi) * HH * WW];
      const int base = (ty + dyi) * 28 + (tx + dxi);
#pragma unroll
      for (int o = 0; o < OBJ; ++o)
        acc[o] = fmaf(wv, msk_s[o * MPAD + base], acc[o]);
    }
  }
#pragma unroll
  for (int o = 0; o < OBJ; ++o)
    out[(((size_t)b * OBJ + o) * HH + y) * WW + x] = acc[o];
}

// ======================================================================
extern "C" void kernel_launch(void* const* d_in, const int* in_sizes, int n_in,
                              void* d_out, int out_size, void* d_ws, size_t ws_size,
                              hipStream_t stream) {
  (void)in_sizes; (void)n_in; (void)out_size; (void)ws_size;
  const float* feat_mem   = (const float*)d_in[0];  // (4,64,64,64)
  const float* feat_query = (const float*)d_in[1];  // (4,64,64,64)
  const float* msk_mem    = (const float*)d_in[2];  // (4,16,256,256)
  const float* feat_flo   = (const float*)d_in[3];  // (1,169,64,64)
  float* out  = (float*)d_out;                      // (4,16,64,64)
  float* wbuf = (float*)d_ws;                       // B*P2*H*W floats (~11 MB)

  corr_wmma_kernel<<<dim3(WW / 16, HH, BB), 128, 0, stream>>>(
      feat_mem, feat_query, wbuf);
  softmax_flo_kernel<<<(BB * HH * WW + 255) / 256, 256, 0, stream>>>(
      feat_flo, wbuf);
  aggregate_kernel<<<dim3(WW / 16, HH / 16, BB), 256, 0, stream>>>(
      msk_mem, wbuf, out);
}